// Neg_Pearson_25726854103345
// MI455X (gfx1250) — compile-verified
//
#include <hip/hip_runtime.h>

typedef float v2f __attribute__((ext_vector_type(2)));
typedef float v4f __attribute__((ext_vector_type(4)));
typedef float v8f __attribute__((ext_vector_type(8)));

// Sum-reduce 64 f32 values (A: 16x4 striped over the wave) into a 16x16 f32
// accumulator using B = ones.  D[m,n] = sum_k A[m,k] + C[m,n]; all 16 columns
// replicate the same row-sums, so total = (sum of all 256 entries) / 16.
__device__ __forceinline__ v8f wmma_sum_acc(v2f a, v8f c) {
    const v2f ones = {1.0f, 1.0f};
    // 8 args: (neg_a, A, neg_b, B, c_mod, C, reuse_a, reuse_b)
    return __builtin_amdgcn_wmma_f32_16x16x4_f32(
        false, a, false, ones, (short)0, c, false, false);
}

__device__ __forceinline__ float hsum8(v8f v) {
    return ((v.s0 + v.s1) + (v.s2 + v.s3)) + ((v.s4 + v.s5) + (v.s6 + v.s7));
}

// One workgroup (256 threads = 8 waves) per row of N=16384 contiguous floats.
// Each wave streams 2048 elements: 16 iterations x 32 lanes x float4.
__global__ __launch_bounds__(256)
void neg_pearson_rows_kernel(const float* __restrict__ preds,
                             const float* __restrict__ labels,
                             float* __restrict__ row_out) {
    const int row  = blockIdx.x;
    const int lane = threadIdx.x & 31;
    const int wave = threadIdx.x >> 5;

    // float4-granular base: row*16384/4 + wave*2048/4 + lane
    const v4f* __restrict__ xp =
        (const v4f*)preds  + (size_t)row * 4096 + (size_t)(wave * 512 + lane);
    const v4f* __restrict__ yp =
        (const v4f*)labels + (size_t)row * 4096 + (size_t)(wave * 512 + lane);

    v8f ax  = {};  // sum x
    v8f ay  = {};  // sum y
    v8f axy = {};  // sum x*y
    v8f axx = {};  // sum x*x
    v8f ayy = {};  // sum y*y

#pragma unroll
    for (int i = 0; i < 16; ++i) {
        // Streaming data, zero reuse, working set >> L2: non-temporal loads.
        v4f x = __builtin_nontemporal_load(xp + i * 32);
        v4f y = __builtin_nontemporal_load(yp + i * 32);

        // Fold 4 elements -> 2 per statistic (sum is associative), then one
        // WMMA accumulates 128 source elements per wave-iteration per stat.
        v2f fx  = { x.x + x.z,                    x.y + x.w };
        v2f fy  = { y.x + y.z,                    y.y + y.w };
        v2f fxy = { fmaf(x.z, y.z, x.x * y.x),    fmaf(x.w, y.w, x.y * y.y) };
        v2f fxx = { fmaf(x.z, x.z, x.x * x.x),    fmaf(x.w, x.w, x.y * x.y) };
        v2f fyy = { fmaf(y.z, y.z, y.x * y.x),    fmaf(y.w, y.w, y.y * y.y) };

        ax  = wmma_sum_acc(fx,  ax);
        ay  = wmma_sum_acc(fy,  ay);
        axy = wmma_sum_acc(fxy, axy);
        axx = wmma_sum_acc(fxx, axx);
        ayy = wmma_sum_acc(fyy, ayy);
    }

    // Per-lane partial (8 accumulator rows each), then wave32 xor-reduction.
    float sx  = hsum8(ax);
    float sy  = hsum8(ay);
    float sxy = hsum8(axy);
    float sxx = hsum8(axx);
    float syy = hsum8(ayy);

#pragma unroll
    for (int m = 16; m >= 1; m >>= 1) {
        sx  += __shfl_xor(sx,  m, 32);
        sy  += __shfl_xor(sy,  m, 32);
        sxy += __shfl_xor(sxy, m, 32);
        sxx += __shfl_xor(sxx, m, 32);
        syy += __shfl_xor(syy, m, 32);
    }

    __shared__ float red[8 * 5];
    if (lane == 0) {
        red[wave * 5 + 0] = sx;
        red[wave * 5 + 1] = sy;
        red[wave * 5 + 2] = sxy;
        red[wave * 5 + 3] = sxx;
        red[wave * 5 + 4] = syy;
    }
    __syncthreads();

    if (threadIdx.x == 0) {
        float tx = 0.f, ty = 0.f, txy = 0.f, txx = 0.f, tyy = 0.f;
#pragma unroll
        for (int w = 0; w < 8; ++w) {
            tx  += red[w * 5 + 0];
            ty  += red[w * 5 + 1];
            txy += red[w * 5 + 2];
            txx += red[w * 5 + 3];
            tyy += red[w * 5 + 4];
        }
        // Undo the 16x column replication of the WMMA ones-trick (exact).
        const float inv16 = 0.0625f;
        tx *= inv16; ty *= inv16; txy *= inv16; txx *= inv16; tyy *= inv16;

        const float Nf  = 16384.0f;
        const float num = Nf * txy - tx * ty;
        const float d1  = Nf * txx - tx * tx;
        const float d2  = Nf * tyy - ty * ty;
        const float p   = num / sqrtf(d1 * d2);
        const float per = (p >= 0.0f) ? (1.0f - p * p) : (1.0f + p * p);
        row_out[row] = per;
    }
}

__global__ __launch_bounds__(256)
void neg_pearson_mean_kernel(const float* __restrict__ row_vals,
                             float* __restrict__ out, int rows) {
    __shared__ float lds[256];
    float s = 0.f;
    for (int i = threadIdx.x; i < rows; i += 256) s += row_vals[i];
    lds[threadIdx.x] = s;
    __syncthreads();
#pragma unroll
    for (int off = 128; off > 0; off >>= 1) {
        if (threadIdx.x < off) lds[threadIdx.x] += lds[threadIdx.x + off];
        __syncthreads();
    }
    if (threadIdx.x == 0) out[0] = lds[0] / (float)rows;
}

extern "C" void kernel_launch(void* const* d_in, const int* in_sizes, int n_in,
                              void* d_out, int out_size, void* d_ws, size_t ws_size,
                              hipStream_t stream) {
    const float* preds  = (const float*)d_in[0];
    const float* labels = (const float*)d_in[1];
    float* out = (float*)d_out;
    float* ws  = (float*)d_ws;

    const int N    = 16384;
    const int rows = in_sizes[0] / N;  // 512 * 4 = 2048

    neg_pearson_rows_kernel<<<rows, 256, 0, stream>>>(preds, labels, ws);
    neg_pearson_mean_kernel<<<1, 256, 0, stream>>>(ws, out, rows);
}